// BlurDegradation_1133871366349
// MI455X (gfx1250) — compile-verified
//
#include <hip/hip_runtime.h>
#include <math.h>

// ---------------------------------------------------------------------------
// BlurDegradation on MI455X (gfx1250):
//   out[b] = (circular 11x11 Gaussian blur)^{t[b]} (x0[b]),  t in [0,20]
//
// Blur is separable & circular => t repeated blurs == ONE separable pass pair
// with the t-fold self-convolved 1D kernel (length 10t+1 <= 201). Each 1D
// circular conv is a banded circulant matmul on the FP32 matrix path
// (V_WMMA_F32_16X16X4_F32 chains, dual accumulators, scalar SALU K-loop).
// LDS staging uses CDNA5 GLOBAL_LOAD_ASYNC_TO_LDS_B32 + s_wait_asynccnt.
// ---------------------------------------------------------------------------

typedef __attribute__((ext_vector_type(2))) float v2f;
typedef __attribute__((ext_vector_type(8))) float v8f;

#define IMG 512
#define NT 20
#define WLEN 256        // padded composed-kernel length, center at 128
#define P1_STRIDE 772   // pass1 LDS row stride (cols -128..639 + pad), bank-staggered
#define P2_STRIDE 48    // pass2 LDS row stride (32-col band), bank-staggered
#define P2_ROWS 280     // >= 64 + 2*5*20 + 3 + 12 halo-extension

// ----- CDNA5 async global->LDS path (guarded: falls back to flat copy) -----
#if defined(__gfx1250__) && __has_builtin(__builtin_amdgcn_global_load_async_to_lds_b32)
#define ASYNC_LDS 1
typedef __attribute__((address_space(1))) int gint_as;
typedef __attribute__((address_space(3))) int lint_as;
#else
#define ASYNC_LDS 0
#endif

__device__ __forceinline__ void lds_copy_b32(float* ldst, const float* gsrc) {
#if ASYNC_LDS
  __builtin_amdgcn_global_load_async_to_lds_b32((gint_as*)gsrc, (lint_as*)ldst, 0, 0);
#else
  *ldst = *gsrc;
#endif
}

__device__ __forceinline__ void lds_copy_fence() {
#if ASYNC_LDS
#if __has_builtin(__builtin_amdgcn_s_wait_asynccnt)
  __builtin_amdgcn_s_wait_asynccnt(0);
#else
  asm volatile("s_wait_asynccnt 0" ::: "memory");
#endif
#endif
}

// ---------------------------------------------------------------------------
// Prep: recover 1D gaussian g from k2d (row 5 / sqrt(center)), then compose
// W[t] = g^{*t} (zero-padded to 256, center 128).  W[0] = delta.
// ---------------------------------------------------------------------------
__global__ __launch_bounds__(256)
void blur_prep_kernel(const float* __restrict__ kernels, float* __restrict__ W) {
  __shared__ float g[16];
  __shared__ float wa[WLEN];
  __shared__ float wb[WLEN];
  const int tid = threadIdx.x;
  if (tid < 11) g[tid] = kernels[5 * 11 + tid] / sqrtf(kernels[5 * 11 + 5]);
  wa[tid] = (tid == 128) ? 1.0f : 0.0f;
  __syncthreads();
  W[tid] = wa[tid];
  float* cur = wa;
  float* nxt = wb;
  for (int s = 1; s <= NT; ++s) {
    float acc = 0.0f;
#pragma unroll
    for (int i = 0; i < 11; ++i) {
      const int j = tid - i + 5;
      if (j >= 0 && j < WLEN) acc += g[i] * cur[j];
    }
    nxt[tid] = acc;
    W[s * WLEN + tid] = acc;
    __syncthreads();
    float* tp = cur; cur = nxt; nxt = tp;
  }
}

// ---------------------------------------------------------------------------
// Pass 1: horizontal circular conv.  Block = 256 thr (8 waves) handles one
// (plane, 16-row band); LDS caches the rows with +/-128 circular halo.
// Weights stored REVERSED so the B fragment is an ascending contiguous pair.
// ---------------------------------------------------------------------------
__global__ __launch_bounds__(256)
void blur_pass1_kernel(const float* __restrict__ x0, const int* __restrict__ t,
                       const float* __restrict__ W, float* __restrict__ tmp) {
  __shared__ float simg[16 * P1_STRIDE];  // cols -128..639 at offset +128
  __shared__ float swr[WLEN];             // swr[j] = w[255-j]

  const int plane = blockIdx.x >> 5;
  const int ytile = blockIdx.x & 31;
  const int b = plane / 3;
  const int tb = __builtin_amdgcn_readfirstlane(t[b]);   // block-uniform -> SGPR
  const int y0 = ytile << 4;
  const float* __restrict__ src = x0 + (size_t)plane * (IMG * IMG);
  float* __restrict__ dst = tmp + (size_t)plane * (IMG * IMG);

  const int tid = threadIdx.x;
  swr[tid] = W[tb * WLEN + (WLEN - 1 - tid)];
  for (int i = tid; i < 16 * 768; i += 256) {
    const int r = i / 768;
    const int col = i - r * 768;                       // -> src col (col-128) mod 512
    lds_copy_b32(&simg[r * P1_STRIDE + col], &src[(y0 + r) * IMG + ((col + 384) & 511)]);
  }
  lds_copy_fence();
  __syncthreads();

  const int lane = tid & 31;
  const int wave = tid >> 5;
  const int m = lane & 15;
  const int kh = lane >> 4;

  for (int xt = wave; xt < 32; xt += 8) {
    const int xb = xt << 4;
    const int lo4 = __builtin_amdgcn_readfirstlane((xb - 5 * tb) & ~3);
    const int nch = __builtin_amdgcn_readfirstlane((xb + 15 + 5 * tb - lo4 + 4) >> 2);
    const int nit = (nch + 1) >> 1;                    // unroll-2; pad chunk hits w==0
    v8f acc0 = {0.f, 0.f, 0.f, 0.f, 0.f, 0.f, 0.f, 0.f};
    v8f acc1 = {0.f, 0.f, 0.f, 0.f, 0.f, 0.f, 0.f, 0.f};
    int c0 = lo4;
    // base LDS/weight indices for this wave's lane role
    const int arow = m * P1_STRIDE + 2 * kh + 128;     // A: image rows
    const int jbase = 127 - (xb + m) + 2 * kh;         // B: reversed weights
    for (int it = 0; it < nit; ++it, c0 += 8) {
      v2f a0, b0, a1, b1;
      a0.x = simg[arow + c0];       a0.y = simg[arow + c0 + 1];
      b0.x = swr[jbase + c0];       b0.y = swr[jbase + c0 + 1];
      a1.x = simg[arow + c0 + 4];   a1.y = simg[arow + c0 + 5];
      b1.x = swr[jbase + c0 + 4];   b1.y = swr[jbase + c0 + 5];
      acc0 = __builtin_amdgcn_wmma_f32_16x16x4_f32(false, a0, false, b0,
                                                   (short)0, acc0, false, false);
      acc1 = __builtin_amdgcn_wmma_f32_16x16x4_f32(false, a1, false, b1,
                                                   (short)0, acc1, false, false);
    }
    const v8f acc = acc0 + acc1;
#pragma unroll
    for (int r = 0; r < 8; ++r) {
      dst[(y0 + (kh << 3) + r) * IMG + xb + m] = acc[r];
    }
  }
}

// ---------------------------------------------------------------------------
// Pass 2: vertical circular conv.  Block = 256 thr handles a 64-row x 32-col
// panel; LDS caches the panel + vertical halo (+12 rows so padded K-chunks
// read initialized data: WMMA propagates NaN even under zero weights).
// ---------------------------------------------------------------------------
__global__ __launch_bounds__(256)
void blur_pass2_kernel(const float* __restrict__ tmp, const int* __restrict__ t,
                       const float* __restrict__ W, float* __restrict__ out) {
  __shared__ float st[P2_ROWS * P2_STRIDE];
  __shared__ float sw[WLEN];

  const int plane = blockIdx.z;
  const int yband = blockIdx.y;           // 0..7  (64 rows)
  const int xband = blockIdx.x;           // 0..15 (32 cols)
  const int b = plane / 3;
  const int tb = __builtin_amdgcn_readfirstlane(t[b]);
  const int ybase = yband << 6;
  const int xbase = xband << 5;
  const float* __restrict__ src = tmp + (size_t)plane * (IMG * IMG);
  float* __restrict__ dst = out + (size_t)plane * (IMG * IMG);

  const int tid = threadIdx.x;
  sw[tid] = W[tb * WLEN + tid];
  const int clo = __builtin_amdgcn_readfirstlane((ybase - 5 * tb) & ~3);
  const int Rl = __builtin_amdgcn_readfirstlane(ybase + 63 + 5 * tb - clo + 13);
  for (int i = tid; i < Rl * 32; i += 256) {
    const int r = i >> 5;
    const int col = i & 31;
    lds_copy_b32(&st[r * P2_STRIDE + col], &src[((clo + r) & 511) * IMG + xbase + col]);
  }
  lds_copy_fence();
  __syncthreads();

  const int lane = tid & 31;
  const int wave = tid >> 5;
  const int m = lane & 15;
  const int kh = lane >> 4;

  {
    const int yt = wave >> 1;
    const int xt = wave & 1;
    const int y0 = ybase + (yt << 4);
    const int xo = xt << 4;
    const int lo4 = __builtin_amdgcn_readfirstlane((y0 - 5 * tb) & ~3);   // >= clo
    const int nch = __builtin_amdgcn_readfirstlane((y0 + 15 + 5 * tb - lo4 + 4) >> 2);
    const int nit = (nch + 1) >> 1;
    v8f acc0 = {0.f, 0.f, 0.f, 0.f, 0.f, 0.f, 0.f, 0.f};
    v8f acc1 = {0.f, 0.f, 0.f, 0.f, 0.f, 0.f, 0.f, 0.f};
    int c0 = lo4;
    const int wbase = 2 * kh - (y0 + m) + 128;          // A: weights (ascending pair)
    const int rbase = 2 * kh - clo;                     // B: image rows in LDS
    for (int it = 0; it < nit; ++it, c0 += 8) {
      v2f a0, b0, a1, b1;
      a0.x = sw[wbase + c0];      a0.y = sw[wbase + c0 + 1];
      a1.x = sw[wbase + c0 + 4];  a1.y = sw[wbase + c0 + 5];
      const int r0 = (rbase + c0) * P2_STRIDE + xo + m;
      b0.x = st[r0];                   b0.y = st[r0 + P2_STRIDE];
      b1.x = st[r0 + 4 * P2_STRIDE];   b1.y = st[r0 + 5 * P2_STRIDE];
      acc0 = __builtin_amdgcn_wmma_f32_16x16x4_f32(false, a0, false, b0,
                                                   (short)0, acc0, false, false);
      acc1 = __builtin_amdgcn_wmma_f32_16x16x4_f32(false, a1, false, b1,
                                                   (short)0, acc1, false, false);
    }
    const v8f acc = acc0 + acc1;
#pragma unroll
    for (int r = 0; r < 8; ++r) {
      dst[(y0 + (kh << 3) + r) * IMG + xbase + xo + m] = acc[r];
    }
  }
}

// ---------------------------------------------------------------------------
// Launch: prep -> row pass (x0 -> ws tmp) -> col pass (tmp -> out)
// Workspace: [0, 21*256 floats) composed kernels; tmp at +8192 floats.
// ---------------------------------------------------------------------------
extern "C" void kernel_launch(void* const* d_in, const int* in_sizes, int n_in,
                              void* d_out, int out_size, void* d_ws, size_t ws_size,
                              hipStream_t stream) {
  const float* x0      = (const float*)d_in[0];
  const int*   tsteps  = (const int*)d_in[1];
  const float* kernels = (const float*)d_in[2];
  float* out = (float*)d_out;

  float* W   = (float*)d_ws;
  float* tmp = (float*)d_ws + 8192;

  blur_prep_kernel<<<1, 256, 0, stream>>>(kernels, W);
  blur_pass1_kernel<<<dim3(192 * 32), 256, 0, stream>>>(x0, tsteps, W, tmp);
  blur_pass2_kernel<<<dim3(16, 8, 192), 256, 0, stream>>>(tmp, tsteps, W, out);
}